// FCOS_73770358276201
// MI455X (gfx1250) — compile-verified
//
#include <hip/hip_runtime.h>
#include <math.h>

// ---------------- problem constants (from reference) ----------------
#define BATCH   16
#define NCLS    20
#define NGT     64
#define NP3     16384   // 128*128, stride 8,  bounds (0, 64)
#define NP4     4096    // 64*64,   stride 16, bounds (64, 128)
#define NP5     1024    // 32*32,   stride 32, bounds (128, inf)
#define NTOT    21504   // NP3+NP4+NP5
#define TOPKN   1000
#define MAXDET  100
#define IOU_THR 0.5f
#define SCORE_THR 0.3f
#define CAND_CAP 2048
#define BLKS_PER_B (NTOT / 256)   // 84

__device__ __forceinline__ float sigf(float x) { return 1.0f / (1.0f + expf(-x)); }

// Map a float to a u32 whose unsigned order equals float order (all floats).
__device__ __forceinline__ unsigned ordkey(float f) {
    unsigned u = __float_as_uint(f);
    return (u & 0x80000000u) ? ~u : (u | 0x80000000u);
}

// =====================================================================
// K1: target assignment. One thread per (batch, location).
// GT boxes (64x5 = 320 floats) staged into LDS with CDNA5 async copy.
// =====================================================================
__global__ void k_targets(const float* __restrict__ locs3,
                          const float* __restrict__ locs4,
                          const float* __restrict__ locs5,
                          const float* __restrict__ gt,     // B*64*5
                          float* __restrict__ out_deltas,   // B*NTOT*4
                          float* __restrict__ out_ctr)      // B*NTOT
{
    __shared__ float sgt[NGT * 5];
    const int t   = threadIdx.x;
    const int b   = blockIdx.x / BLKS_PER_B;
    const int loc = (blockIdx.x % BLKS_PER_B) * 256 + t;
    const float* gsrc = gt + (size_t)b * NGT * 5;

#if defined(__gfx1250__) && __has_builtin(__builtin_amdgcn_global_load_async_to_lds_b32)
    if (t < NGT * 5) {
        // builtin expects non-const AS1 pointer; C-style cast strips const +
        // performs the (no-op) generic->global addrspacecast. LDS side is a
        // genuine generic->AS3 addrspacecast of the __shared__ pointer.
        __builtin_amdgcn_global_load_async_to_lds_b32(
            (__attribute__((address_space(1))) int*)(gsrc + t),
            (__attribute__((address_space(3))) int*)(&sgt[t]),
            0, 0);
    }
#if __has_builtin(__builtin_amdgcn_s_wait_asynccnt)
    __builtin_amdgcn_s_wait_asynccnt(0);
#else
    asm volatile("s_wait_asynccnt 0" ::: "memory");
#endif
#else
    if (t < NGT * 5) sgt[t] = gsrc[t];
#endif
    __syncthreads();

    float c0, c1, stride, lo, hi;
    if (loc < NP3)              { c0 = locs3[loc * 2]; c1 = locs3[loc * 2 + 1]; stride = 8.f;  lo = 0.f;   hi = 64.f; }
    else if (loc < NP3 + NP4)   { int li = loc - NP3;       c0 = locs4[li * 2]; c1 = locs4[li * 2 + 1]; stride = 16.f; lo = 64.f;  hi = 128.f; }
    else                        { int li = loc - NP3 - NP4; c0 = locs5[li * 2]; c1 = locs5[li * 2 + 1]; stride = 32.f; lo = 128.f; hi = INFINITY; }

    // argmax over 64 GT boxes; first-max semantics via strict '>'
    float bestq = -INFINITY;
    int   bidx  = 0;
    for (int j = 0; j < NGT; ++j) {
        float x0 = sgt[j * 5 + 0], y0 = sgt[j * 5 + 1];
        float x1 = sgt[j * 5 + 2], y1 = sgt[j * 5 + 3];
        float l  = c0 - x0, tt = c1 - y0, r = x1 - c0, bb = y1 - c1;
        float mn = fminf(fminf(l, tt), fminf(r, bb));
        float mx = fmaxf(fmaxf(l, tt), fmaxf(r, bb));
        bool  ok = (mn > 0.f) && (mx > lo) && (mx < hi);
        float area = (x1 - x0) * (y1 - y0);
        float q = ok ? (1e8f - area) : 0.f;
        if (q > bestq) { bestq = q; bidx = j; }
    }

    const size_t o  = (size_t)b * NTOT + loc;
    const bool   bg = (bestq < 1e-5f);   // idx = -1 in reference
    if (bg) {
        out_deltas[o * 4 + 0] = -1.f; out_deltas[o * 4 + 1] = -1.f;
        out_deltas[o * 4 + 2] = -1.f; out_deltas[o * 4 + 3] = -1.f;
        out_ctr[o] = -1.f;
    } else {
        float m0 = sgt[bidx * 5 + 0], m1 = sgt[bidx * 5 + 1];
        float m2 = sgt[bidx * 5 + 2], m3 = sgt[bidx * 5 + 3];
        float d0 = (c0 - m0) / stride, d1 = (c1 - m1) / stride;
        float d2 = (m2 - c0) / stride, d3 = (m3 - c1) / stride;
        float num = fminf(d0, d2) * fminf(d1, d3);
        float den = fmaxf(d0, d2) * fmaxf(d1, d3);
        out_deltas[o * 4 + 0] = d0; out_deltas[o * 4 + 1] = d1;
        out_deltas[o * 4 + 2] = d2; out_deltas[o * 4 + 3] = d3;
        out_ctr[o] = sqrtf(num / den);
    }
}

// =====================================================================
// K2: decode boxes + per-class scores (max / first-argmax over C=20).
// =====================================================================
__global__ void k_decode(const float* __restrict__ locs3,
                         const float* __restrict__ locs4,
                         const float* __restrict__ locs5,
                         const float* __restrict__ cls3, const float* __restrict__ reg3, const float* __restrict__ ctr3,
                         const float* __restrict__ cls4, const float* __restrict__ reg4, const float* __restrict__ ctr4,
                         const float* __restrict__ cls5, const float* __restrict__ reg5, const float* __restrict__ ctr5,
                         float* __restrict__ ws_boxes, float* __restrict__ ws_score, int* __restrict__ ws_cls)
{
    const int b   = blockIdx.y;
    const int loc = blockIdx.x * 256 + threadIdx.x;

    const float* cls; const float* reg; float ctrv, c0, c1, stride;
    if (loc < NP3) {
        cls = cls3 + ((size_t)b * NP3 + loc) * NCLS;
        reg = reg3 + ((size_t)b * NP3 + loc) * 4;
        ctrv = ctr3[(size_t)b * NP3 + loc];
        c0 = locs3[loc * 2]; c1 = locs3[loc * 2 + 1]; stride = 8.f;
    } else if (loc < NP3 + NP4) {
        int li = loc - NP3;
        cls = cls4 + ((size_t)b * NP4 + li) * NCLS;
        reg = reg4 + ((size_t)b * NP4 + li) * 4;
        ctrv = ctr4[(size_t)b * NP4 + li];
        c0 = locs4[li * 2]; c1 = locs4[li * 2 + 1]; stride = 16.f;
    } else {
        int li = loc - NP3 - NP4;
        cls = cls5 + ((size_t)b * NP5 + li) * NCLS;
        reg = reg5 + ((size_t)b * NP5 + li) * 4;
        ctrv = ctr5[(size_t)b * NP5 + li];
        c0 = locs5[li * 2]; c1 = locs5[li * 2 + 1]; stride = 32.f;
    }
    __builtin_prefetch(cls, 0, 1);   // global_prefetch_b8 on gfx1250

    const float sctr = sigf(ctrv);
    float best = -INFINITY; int bestc = 0;
    for (int c = 0; c < NCLS; ++c) {
        float s = sqrtf(sigf(cls[c]) * sctr);
        if (s > best) { best = s; bestc = c; }
    }
    float d0 = fmaxf(reg[0], 0.f), d1 = fmaxf(reg[1], 0.f);
    float d2 = fmaxf(reg[2], 0.f), d3 = fmaxf(reg[3], 0.f);

    const size_t o = (size_t)b * NTOT + loc;
    float4 bx = make_float4(c0 - d0 * stride, c1 - d1 * stride,
                            c0 + d2 * stride, c1 + d3 * stride);
    *(float4*)(ws_boxes + o * 4) = bx;
    ws_score[o] = best;
    ws_cls[o]   = bestc;
}

// =====================================================================
// K3: per-batch radix select of the exact 1000th-largest score
// (scores are positive floats -> u32 bit pattern is order-preserving).
// =====================================================================
__global__ void k_select(const float* __restrict__ ws_score,
                         unsigned* __restrict__ cutoff,
                         unsigned* __restrict__ candcnt)
{
    __shared__ unsigned hist[256];
    __shared__ unsigned s_prefix;
    __shared__ int      s_k;
    const int b = blockIdx.x, t = threadIdx.x;
    if (t == 0) { s_prefix = 0u; s_k = TOPKN; candcnt[b] = 0u; }
    const float* sc = ws_score + (size_t)b * NTOT;

    for (int pass = 0; pass < 4; ++pass) {
        const int shift = 24 - 8 * pass;
        hist[t] = 0u;
        __syncthreads();
        const unsigned prefix = s_prefix;
        const unsigned pmask  = (pass == 0) ? 0u : (0xFFFFFFFFu << (shift + 8));
        for (int i = t; i < NTOT; i += 256) {
            unsigned key = __float_as_uint(sc[i]);
            if ((key & pmask) == prefix)
                atomicAdd(&hist[(key >> shift) & 0xFF], 1u);
        }
        __syncthreads();
        if (t == 0) {
            int k = s_k, bsel = 0;
            for (int bb = 255; bb >= 0; --bb) {
                int h = (int)hist[bb];
                if (k - h <= 0) { bsel = bb; break; }
                k -= h;
            }
            s_prefix = prefix | ((unsigned)bsel << shift);
            s_k = k;
        }
        __syncthreads();
    }
    if (t == 0) cutoff[b] = s_prefix;
}

// =====================================================================
// K4: gather all candidates with key >= cutoff (unordered).
// =====================================================================
__global__ void k_compact(const float* __restrict__ ws_score,
                          const unsigned* __restrict__ cutoff,
                          unsigned* __restrict__ candcnt,
                          unsigned* __restrict__ cand_key,
                          unsigned* __restrict__ cand_idx)
{
    const int b = blockIdx.y;
    const int i = blockIdx.x * 256 + threadIdx.x;
    unsigned key = __float_as_uint(ws_score[(size_t)b * NTOT + i]);
    if (key >= cutoff[b]) {
        unsigned p = atomicAdd(&candcnt[b], 1u);
        if (p < CAND_CAP) {
            cand_key[(size_t)b * CAND_CAP + p] = key;
            cand_idx[(size_t)b * CAND_CAP + p] = (unsigned)i;
        }
    }
}

// =====================================================================
// K5: exact rank among candidates (score desc, index asc — jax.lax.top_k
// stable order); scatter rank < 1000 entries into top-k slot arrays.
// =====================================================================
__global__ void k_rank(const unsigned* __restrict__ candcnt,
                       const unsigned* __restrict__ cand_key,
                       const unsigned* __restrict__ cand_idx,
                       const float* __restrict__ ws_boxes,
                       const float* __restrict__ ws_score,
                       const int* __restrict__ ws_cls,
                       float* __restrict__ topk_box,
                       float* __restrict__ topk_score,
                       int* __restrict__ topk_cls)
{
    __shared__ unsigned skey[CAND_CAP];
    __shared__ unsigned sidx[CAND_CAP];
    const int b = blockIdx.x, t = threadIdx.x;
    int n = (int)candcnt[b];
    if (n > CAND_CAP) n = CAND_CAP;

    for (int i = t; i < n; i += 256) {
        skey[i] = cand_key[(size_t)b * CAND_CAP + i];
        sidx[i] = cand_idx[(size_t)b * CAND_CAP + i];
    }
    __syncthreads();

    for (int i = t; i < n; i += 256) {
        const unsigned ki = skey[i], xi = sidx[i];
        int rank = 0;
        for (int j = 0; j < n; ++j) {
            unsigned kj = skey[j];
            rank += (kj > ki) || (kj == ki && sidx[j] < xi);
        }
        if (rank < TOPKN) {
            const size_t src = (size_t)b * NTOT + xi;
            const size_t dst = (size_t)b * TOPKN + rank;
            topk_score[dst] = ws_score[src];
            topk_cls[dst]   = ws_cls[src];
            *(float4*)(topk_box + dst * 4) = *(const float4*)(ws_boxes + src * 4);
        }
    }
}

// =====================================================================
// K6: sequential NMS, one 1024-thread block per batch; thread i owns
// top-k row i. Packed-u64 argmax (tie -> lowest index). Writes dets.
// =====================================================================
__global__ void k_nms(const float* __restrict__ topk_box,
                      const float* __restrict__ topk_score,
                      const int* __restrict__ topk_cls,
                      float* __restrict__ out_dets)   // B*TOPKN*5
{
    __shared__ float sb0[TOPKN], sb1[TOPKN], sb2[TOPKN], sb3[TOPKN], sarea[TOPKN];
    __shared__ unsigned long long red[1024];
    __shared__ float fred[1024];

    const int b = blockIdx.x, t = threadIdx.x;
    const bool have = (t < TOPKN);

    float bx0 = 0.f, bx1 = 0.f, bx2 = 0.f, bx3 = 0.f, sc = -INFINITY;
    int cid = 0;
    if (have) {
        const float* p = topk_box + ((size_t)b * TOPKN + t) * 4;
        bx0 = p[0]; bx1 = p[1]; bx2 = p[2]; bx3 = p[3];
        sc  = topk_score[(size_t)b * TOPKN + t];
        cid = topk_cls[(size_t)b * TOPKN + t];
    }
    const bool valid = have && (sc > SCORE_THR);

    // max_coord = max over where(valid, bx, 0)
    fred[t] = valid ? fmaxf(fmaxf(bx0, bx1), fmaxf(bx2, bx3))
                    : (have ? 0.0f : -INFINITY);
    __syncthreads();
    for (int s = 512; s > 0; s >>= 1) {
        if (t < s) fred[t] = fmaxf(fred[t], fred[t + s]);
        __syncthreads();
    }
    const float mc  = fred[0];
    const float off = (float)cid * (mc + 1.0f);

    const float n0 = bx0 + off, n1 = bx1 + off, n2 = bx2 + off, n3 = bx3 + off;
    const float ar = (n2 - n0) * (n3 - n1);
    if (have) { sb0[t] = n0; sb1[t] = n1; sb2[t] = n2; sb3[t] = n3; sarea[t] = ar; }

    float smask = valid ? sc : -INFINITY;
    bool  keep  = false;
    __syncthreads();

    for (int it = 0; it < MAXDET; ++it) {
        unsigned long long pk =
            ((unsigned long long)ordkey(smask) << 32) |
            (unsigned long long)(0xFFFFFFFFu - (unsigned)t);
        red[t] = have ? pk : 0ull;
        __syncthreads();
        for (int s = 512; s > 0; s >>= 1) {
            if (t < s) { unsigned long long o = red[t + s]; if (o > red[t]) red[t] = o; }
            __syncthreads();
        }
        const unsigned long long top = red[0];
        __syncthreads();
        if ((unsigned)(top >> 32) < 0x80000000u) break;   // no active (all -inf)

        const int sel = (int)(0xFFFFFFFFu - (unsigned)(top & 0xFFFFFFFFu));
        const float c0 = sb0[sel], c1 = sb1[sel], c2 = sb2[sel], c3 = sb3[sel];
        const float ca = sarea[sel];
        if (t == sel) keep = true;
        if (have) {
            float xx1 = fmaxf(c0, n0), yy1 = fmaxf(c1, n1);
            float xx2 = fminf(c2, n2), yy2 = fminf(c3, n3);
            float inter = fmaxf(xx2 - xx1, 0.f) * fmaxf(yy2 - yy1, 0.f);
            float iou = inter / (ca + ar - inter);
            if (iou > IOU_THR) smask = -INFINITY;
        }
    }

    if (have) {
        float* o = out_dets + ((size_t)b * TOPKN + t) * 5;
        if (keep) { o[0] = bx0; o[1] = bx1; o[2] = bx2; o[3] = bx3; o[4] = sc; }
        else      { o[0] = 0.f; o[1] = 0.f; o[2] = 0.f; o[3] = 0.f; o[4] = 0.f; }
    }
}

// =====================================================================
extern "C" void kernel_launch(void* const* d_in, const int* in_sizes, int n_in,
                              void* d_out, int out_size, void* d_ws, size_t ws_size,
                              hipStream_t stream) {
    // setup_inputs() dict order:
    const float* locs3 = (const float*)d_in[0];
    const float* locs4 = (const float*)d_in[1];
    const float* locs5 = (const float*)d_in[2];
    const float* gt    = (const float*)d_in[3];
    const float* cls3  = (const float*)d_in[4];
    const float* reg3  = (const float*)d_in[5];
    const float* ctr3  = (const float*)d_in[6];
    const float* cls4  = (const float*)d_in[7];
    const float* reg4  = (const float*)d_in[8];
    const float* ctr4  = (const float*)d_in[9];
    const float* cls5  = (const float*)d_in[10];
    const float* reg5  = (const float*)d_in[11];
    const float* ctr5  = (const float*)d_in[12];

    // outputs: dets (B*1000*5) | gt_deltas (B*NTOT*4) | gt_ctr (B*NTOT)
    float* out        = (float*)d_out;
    float* out_dets   = out;
    float* out_deltas = out + (size_t)BATCH * TOPKN * 5;
    float* out_ctr    = out_deltas + (size_t)BATCH * NTOT * 4;

    // workspace carve-up (~8.9 MB, 16B-aligned slices)
    float*    ws_boxes   = (float*)d_ws;                                   // B*NTOT*4
    float*    ws_score   = ws_boxes + (size_t)BATCH * NTOT * 4;            // B*NTOT
    int*      ws_cls     = (int*)(ws_score + (size_t)BATCH * NTOT);        // B*NTOT
    float*    topk_box   = (float*)(ws_cls + (size_t)BATCH * NTOT);        // B*TOPKN*4
    float*    topk_score = topk_box + (size_t)BATCH * TOPKN * 4;           // B*TOPKN
    int*      topk_cls   = (int*)(topk_score + (size_t)BATCH * TOPKN);     // B*TOPKN
    unsigned* cand_key   = (unsigned*)(topk_cls + (size_t)BATCH * TOPKN);  // B*CAND_CAP
    unsigned* cand_idx   = cand_key + (size_t)BATCH * CAND_CAP;            // B*CAND_CAP
    unsigned* cutoff     = cand_idx + (size_t)BATCH * CAND_CAP;            // B
    unsigned* candcnt    = cutoff + BATCH;                                 // B

    k_targets<<<dim3(BLKS_PER_B * BATCH), dim3(256), 0, stream>>>(
        locs3, locs4, locs5, gt, out_deltas, out_ctr);

    k_decode<<<dim3(BLKS_PER_B, BATCH), dim3(256), 0, stream>>>(
        locs3, locs4, locs5,
        cls3, reg3, ctr3, cls4, reg4, ctr4, cls5, reg5, ctr5,
        ws_boxes, ws_score, ws_cls);

    k_select<<<dim3(BATCH), dim3(256), 0, stream>>>(ws_score, cutoff, candcnt);

    k_compact<<<dim3(BLKS_PER_B, BATCH), dim3(256), 0, stream>>>(
        ws_score, cutoff, candcnt, cand_key, cand_idx);

    k_rank<<<dim3(BATCH), dim3(256), 0, stream>>>(
        candcnt, cand_key, cand_idx, ws_boxes, ws_score, ws_cls,
        topk_box, topk_score, topk_cls);

    k_nms<<<dim3(BATCH), dim3(1024), 0, stream>>>(
        topk_box, topk_score, topk_cls, out_dets);
}